// LightGCN_3109556322387
// MI455X (gfx1250) — compile-verified
//
#include <hip/hip_runtime.h>
#include <stdint.h>

typedef float v2f __attribute__((ext_vector_type(2)));
typedef float v8f __attribute__((ext_vector_type(8)));

#define EMB_D 64

// ---------------------------------------------------------------------------
// init: e_cur = acc = concat(user_emb, item_emb), vectorized float4
// ---------------------------------------------------------------------------
__global__ void lgcn_init_kernel(const float4* __restrict__ uemb,
                                 const float4* __restrict__ iemb,
                                 float4* __restrict__ e_cur,
                                 float4* __restrict__ acc,
                                 long long n_user_vec4, long long total_vec4) {
    long long t = (long long)blockIdx.x * blockDim.x + threadIdx.x;
    if (t >= total_vec4) return;
    float4 v = (t < n_user_vec4) ? uemb[t] : iemb[t - n_user_vec4];
    e_cur[t] = v;
    acc[t]   = v;
}

__global__ void lgcn_zero_kernel(float4* __restrict__ p, long long total_vec4) {
    long long t = (long long)blockIdx.x * blockDim.x + threadIdx.x;
    if (t >= total_vec4) return;
    p[t] = make_float4(0.f, 0.f, 0.f, 0.f);
}

__global__ void lgcn_accadd_kernel(float4* __restrict__ acc,
                                   const float4* __restrict__ nxt,
                                   long long total_vec4) {
    long long t = (long long)blockIdx.x * blockDim.x + threadIdx.x;
    if (t >= total_vec4) return;
    float4 a = acc[t], b = nxt[t];
    a.x += b.x; a.y += b.y; a.z += b.z; a.w += b.w;
    acc[t] = a;
}

// ---------------------------------------------------------------------------
// SpMM scatter: one wave (32 lanes) per edge, 2 floats/lane => coalesced
// 256B gather + 256B hardware-f32-atomic scatter per edge.
// ---------------------------------------------------------------------------
__global__ void lgcn_spmm_kernel(const float* __restrict__ adj_data,
                                 const long long* __restrict__ adj_idx,
                                 const float* __restrict__ e_cur,
                                 float* __restrict__ e_next,
                                 long long nnz) {
    const int lane = threadIdx.x & 31;
    const long long wave   = (long long)blockIdx.x * (blockDim.x >> 5) + (threadIdx.x >> 5);
    const long long nwaves = (long long)gridDim.x * (blockDim.x >> 5);
    for (long long e = wave; e < nnz; e += nwaves) {
        long long row = adj_idx[2 * e];
        long long col = adj_idx[2 * e + 1];
        float w = adj_data[e];
        const float* __restrict__ src = e_cur + (col << 6);
        float* __restrict__ dst = e_next + (row << 6);
        float x0 = src[lane];
        float x1 = src[lane + 32];
        unsafeAtomicAdd(dst + lane,      w * x0);   // global_atomic_add_f32
        unsafeAtomicAdd(dst + lane + 32, w * x1);
    }
}

// ---------------------------------------------------------------------------
// Final rowwise dots via V_WMMA_F32_16X16X4_F32: one wave per 16 pairs.
// A(16x4) = user rows, B(4x16) = item rows; D diagonal = dot products, f32.
// Per ISA VGPR layouts, each lane loads row (lane&15) at K offset
// kbase = (lane>>4)*2 for BOTH A and B, as a float2, each K=4 step.
// Diagonal of D (8 VGPRs): p in [0,8) -> VGPR p @ lane p;
//                          p in [8,16) -> VGPR p-8 @ lane p+16.
// ---------------------------------------------------------------------------
__global__ void lgcn_dot_kernel(const float* __restrict__ acc,
                                const long long* __restrict__ user_idx,
                                const long long* __restrict__ item_idx,
                                float* __restrict__ out,
                                long long n_users, int npairs) {
    const int lane = threadIdx.x & 31;
    const int tile = (blockIdx.x * (blockDim.x >> 5)) + (threadIdx.x >> 5);
    if (tile * 16 >= npairs) return;   // wave-uniform: EXEC stays all-ones

    const int p_load = tile * 16 + (lane & 15);
    const long long urow = user_idx[p_load];
    const long long irow = n_users + item_idx[p_load];
    const float* __restrict__ ubase = acc + (urow << 6);
    const float* __restrict__ ibase = acc + (irow << 6);
    const int kbase = (lane >> 4) * 2;

#if __has_builtin(__builtin_amdgcn_wmma_f32_16x16x4_f32)
    v8f c = {0.f, 0.f, 0.f, 0.f, 0.f, 0.f, 0.f, 0.f};
#pragma unroll
    for (int k0 = 0; k0 < EMB_D; k0 += 4) {
        v2f a = *(const v2f*)(ubase + k0 + kbase);
        v2f b = *(const v2f*)(ibase + k0 + kbase);
        c = __builtin_amdgcn_wmma_f32_16x16x4_f32(
                /*neg_a=*/false, a, /*neg_b=*/false, b,
                /*c_mod=*/(short)0, c, /*reuse_a=*/false, /*reuse_b=*/false);
    }
    // extract diagonal
    bool active = false; int j = 0, p = 0;
    if (lane < 8)        { active = true; j = lane;      p = lane; }
    else if (lane >= 24) { active = true; j = lane - 24; p = lane - 16; }
    float val = 0.f;
#pragma unroll
    for (int q = 0; q < 8; ++q) val = (j == q) ? c[q] : val;
    int op = tile * 16 + p;
    if (active && op < npairs) out[op] = val * 0.0625f;   // (1/4)*(1/4)
#else
    // fallback: plain dot (keeps kernel compiling if builtin is absent)
    float val = 0.f;
#pragma unroll
    for (int k = 0; k < EMB_D; ++k) val += ubase[k] * ibase[k];
    int op = p_load;
    if (lane < 16 && op < npairs) out[op] = val * 0.0625f;
#endif
}

// ---------------------------------------------------------------------------
extern "C" void kernel_launch(void* const* d_in, const int* in_sizes, int n_in,
                              void* d_out, int out_size, void* d_ws, size_t ws_size,
                              hipStream_t stream) {
    const float*     user_emb = (const float*)d_in[0];
    const float*     item_emb = (const float*)d_in[1];
    const float*     adj_data = (const float*)d_in[2];
    const long long* adj_idx  = (const long long*)d_in[3];
    const long long* user_idx = (const long long*)d_in[4];
    const long long* item_idx = (const long long*)d_in[5];
    float* out = (float*)d_out;

    const long long n_users = in_sizes[0] / EMB_D;   // 100000
    const long long n_items = in_sizes[1] / EMB_D;   // 50000
    const long long N       = n_users + n_items;     // 150000
    const long long nnz     = in_sizes[2];           // 6.4M
    const int npairs        = in_sizes[4];           // 4096

    const long long bufElems = N * EMB_D;            // 9.6M floats
    float* e_cur  = (float*)d_ws;
    float* e_next = e_cur  + bufElems;
    float* acc    = e_next + bufElems;

    const long long totalVec4 = N * (EMB_D / 4);     // 2.4M float4
    const long long userVec4  = n_users * (EMB_D / 4);
    const int ebBlocks = (int)((totalVec4 + 255) / 256);

    lgcn_init_kernel<<<ebBlocks, 256, 0, stream>>>(
        (const float4*)user_emb, (const float4*)item_emb,
        (float4*)e_cur, (float4*)acc, userVec4, totalVec4);

    const int spmmBlocks = 16384;  // 8 waves/block -> 131072 waves, grid-stride
    for (int layer = 0; layer < 3; ++layer) {
        lgcn_zero_kernel<<<ebBlocks, 256, 0, stream>>>((float4*)e_next, totalVec4);
        lgcn_spmm_kernel<<<spmmBlocks, 256, 0, stream>>>(
            adj_data, adj_idx, e_cur, e_next, nnz);
        lgcn_accadd_kernel<<<ebBlocks, 256, 0, stream>>>(
            (float4*)acc, (const float4*)e_next, totalVec4);
        float* tmp = e_cur; e_cur = e_next; e_next = tmp;
    }

    const int dotWaves  = (npairs + 15) / 16;        // 256 waves
    const int dotBlocks = (dotWaves + 7) / 8;        // 8 waves/block
    lgcn_dot_kernel<<<dotBlocks, 256, 0, stream>>>(
        acc, user_idx, item_idx, out, n_users, npairs);
}